// EdgeTransformerLayer_43585328120339
// MI455X (gfx1250) — compile-verified
//
#include <hip/hip_runtime.h>
#include <hip/hip_bf16.h>
#include <math.h>

// ---------------------------------------------------------------------------
// EdgeTransformer layer for MI455X (gfx1250, wave32, WMMA).
// B=2, N=64, E=256, H=8, dk=32, F=512, R = B*N*N = 8192 rows.
// All GEMMs use V_WMMA_F32_16X16X4_F32 (fp32 in/out: reference is fp32 and the
// kernel is memory/L2 balanced, so fp32 matrix rate is not the bottleneck).
// ---------------------------------------------------------------------------

typedef float v2f __attribute__((ext_vector_type(2)));
typedef float v8f __attribute__((ext_vector_type(8)));

#ifndef __has_builtin
#define __has_builtin(x) 0
#endif
#if __has_builtin(__builtin_amdgcn_wmma_f32_16x16x4_f32)
#define HAVE_WMMA_F32_X4 1
#else
#define HAVE_WMMA_F32_X4 0
#endif

// D = A(16x4 f32) * B(4x16 f32) + C(16x16 f32), whole-wave op.
// Operand layout per CDNA5 ISA 7.12.2:
//   A: lane<16 -> M=lane,    a.x=K0..  ; lane>=16 -> M=lane-16, a.x=K2..
//   B: lane<16 -> N=lane,    b.x=K0,b.y=K1 ; lane>=16 -> b.x=K2,b.y=K3
//   C/D: VGPR r -> M=r (lanes 0-15) / M=r+8 (lanes 16-31), N=lane&15
static __device__ inline v8f wmma_f32x4(v2f a, v2f b, v8f c) {
#if HAVE_WMMA_F32_X4
  return __builtin_amdgcn_wmma_f32_16x16x4_f32(
      /*neg_a=*/false, a, /*neg_b=*/false, b,
      /*c_mod=*/(short)0, c, /*reuse_a=*/false, /*reuse_b=*/false);
#else
  // Compile-safety fallback only (should not be taken on gfx1250).
  c[0] += a.x * b.x + a.y * b.y;
  return c;
#endif
}

static __device__ inline float wave_reduce_add(float v) {
  // wave32: 5 xor steps
  #pragma unroll
  for (int off = 16; off > 0; off >>= 1) v += __shfl_xor(v, off, 32);
  return v;
}

// Wave computes NT adjacent 16x16 output tiles of Y = A(16 rows x K) * B(K x N).
// A: global fp32, row stride lda. B: global fp32, row stride ldb.
template <int NT>
static __device__ inline void gemm_panel(const float* __restrict__ A, int lda,
                                         int rowbase,
                                         const float* __restrict__ Bm, int ldb,
                                         int colbase, int K, v8f* acc) {
  const int lane  = threadIdx.x & 31;
  const int lrow  = lane & 15;
  const int khalf = lane >> 4;  // 0 or 1
  const float* arow = A + (size_t)(rowbase + lrow) * lda + 2 * khalf;
  for (int kb = 0; kb < K; kb += 4) {
    v2f a = *(const v2f*)(arow + kb);
    #pragma unroll
    for (int t = 0; t < NT; ++t) {
      int col = colbase + t * 16 + lrow;
      v2f b;
      b.x = Bm[(size_t)(kb + 2 * khalf) * ldb + col];
      b.y = Bm[(size_t)(kb + 2 * khalf + 1) * ldb + col];
      acc[t] = wmma_f32x4(a, b, acc[t]);
    }
  }
}

// ---------------------------------------------------------------------------
// K1: q/k/v1/v2 projections. blockIdx.y selects which weight/output.
// Block = 256 thr (8 waves) -> 16 rows x 256 cols panel; 2 tiles per wave.
// ---------------------------------------------------------------------------
__global__ void __launch_bounds__(256)
proj_kernel(const float* __restrict__ x, const float* __restrict__ Wq,
            const float* __restrict__ Wk, const float* __restrict__ Wv1,
            const float* __restrict__ Wv2, float* __restrict__ q,
            float* __restrict__ k, float* __restrict__ v1,
            float* __restrict__ v2) {
  const float* W;
  float* Y;
  switch (blockIdx.y) {
    case 0:  W = Wq;  Y = q;  break;
    case 1:  W = Wk;  Y = k;  break;
    case 2:  W = Wv1; Y = v1; break;
    default: W = Wv2; Y = v2; break;
  }
  const int rowbase = blockIdx.x * 16;
  const int wid  = threadIdx.x >> 5;
  const int lane = threadIdx.x & 31;
  const int lrow = lane & 15;
  const int rh   = (lane >> 4) * 8;

  v8f acc[2] = {};
  gemm_panel<2>(x, 256, rowbase, W, 256, wid * 32, 256, acc);

  #pragma unroll
  for (int t = 0; t < 2; ++t) {
    int col = wid * 32 + t * 16 + lrow;
    #pragma unroll
    for (int r = 0; r < 8; ++r)
      Y[(size_t)(rowbase + r + rh) * 256 + col] = acc[t][r];
  }
}

// ---------------------------------------------------------------------------
// K2: scores[b,i,l,j,h] = (1/sqrt(dk)) * sum_d q[b,i,l,h,d]*k[b,l,j,h,d]
// One block per (b,h,l): 64x64x32 GEMM (16 wmma tiles, 2 per wave).
// Stored as att[(bh*64+l)*4096 + i*64 + j].
// ---------------------------------------------------------------------------
__global__ void __launch_bounds__(256)
scores_kernel(const float* __restrict__ q, const float* __restrict__ k,
              float* __restrict__ att) {
  const int idx = blockIdx.x;       // ((b*8+h)*64 + l)
  const int l   = idx & 63;
  const int bh  = idx >> 6;
  const int h   = bh & 7;
  const int b   = bh >> 3;
  const int wid   = threadIdx.x >> 5;
  const int lane  = threadIdx.x & 31;
  const int lrow  = lane & 15;
  const int khalf = lane >> 4;
  const float scale = 0.17677669529663687f;  // 1/sqrt(32)

  // q[b,i,l,h,d] : row (b*4096 + i*64 + l), elem h*32+d
  const float* qb = q + ((size_t)b * 4096 + l) * 256 + h * 32;
  // k[b,l,j,h,d] : row (b*4096 + l*64 + j)
  const float* kb = k + ((size_t)b * 4096 + (size_t)l * 64) * 256 + h * 32;
  float* outp = att + (size_t)idx * 4096;

  #pragma unroll
  for (int s = 0; s < 2; ++s) {
    const int st = wid * 2 + s;
    const int it = st >> 2, jt = st & 3;
    v8f c = {};
    const float* arow = qb + (size_t)(it * 16 + lrow) * (64 * 256) + 2 * khalf;
    const float* brow = kb + (size_t)(jt * 16 + lrow) * 256 + 2 * khalf;
    #pragma unroll
    for (int kk = 0; kk < 32; kk += 4) {
      v2f a  = *(const v2f*)(arow + kk);
      v2f bb = *(const v2f*)(brow + kk);  // contiguous in d
      c = wmma_f32x4(a, bb, c);
    }
    const int j = jt * 16 + lrow;
    #pragma unroll
    for (int r = 0; r < 8; ++r) {
      const int i = it * 16 + r + khalf * 8;
      outp[i * 64 + j] = c[r] * scale;
    }
  }
}

// ---------------------------------------------------------------------------
// K3: softmax over the middle-node axis l (stride 4096 between l's).
// Block = 4 i's x 64 j's; each thread owns one (b,h,i,j), 64 values in VGPRs.
// ---------------------------------------------------------------------------
__global__ void __launch_bounds__(256)
softmax_l_kernel(float* __restrict__ att) {
  const int idx = blockIdx.x;  // (bh*16 + ig)
  const int ig  = idx & 15;
  const int bh  = idx >> 4;
  const int t   = threadIdx.x;
  const int i   = ig * 4 + (t >> 6);
  const int j   = t & 63;
  float* base = att + (size_t)bh * 64 * 4096 + i * 64 + j;

  float vals[64];
  float mx = -3.402823466e38f;
  #pragma unroll
  for (int l = 0; l < 64; ++l) {
    vals[l] = base[(size_t)l * 4096];
    mx = fmaxf(mx, vals[l]);
  }
  float s = 0.f;
  #pragma unroll
  for (int l = 0; l < 64; ++l) {
    vals[l] = __expf(vals[l] - mx);
    s += vals[l];
  }
  const float inv = 1.0f / s;
  #pragma unroll
  for (int l = 0; l < 64; ++l) base[(size_t)l * 4096] = vals[l] * inv;
}

// ---------------------------------------------------------------------------
// K4: out[b,i,j,h,d] = sum_l att[b,i,l,j,h] * v1[b,i,l,h,d] * v2[b,l,j,h,d]
// Elementwise in d -> VALU loop. att(64x64) + v1(64x32) staged in LDS,
// v2 streamed from L2 (prefetch one l ahead).
// ---------------------------------------------------------------------------
__global__ void __launch_bounds__(256)
edge_av_kernel(const float* __restrict__ att, const float* __restrict__ v1,
               const float* __restrict__ v2, float* __restrict__ out_pre) {
  const int idx = blockIdx.x;  // (b*8+h)*64 + i
  const int i   = idx & 63;
  const int bh  = idx >> 6;
  const int h   = bh & 7;
  const int b   = bh >> 3;
  const int tid = threadIdx.x;

  __shared__ float s_att[64 * 64];  // [l][j]
  __shared__ float s_v1[64 * 32];   // [l][d]

  const float* attb = att + (size_t)bh * 64 * 4096 + i * 64;  // + l*4096 + j
  for (int tt = tid; tt < 4096; tt += 256) {
    const int l = tt >> 6, j = tt & 63;
    s_att[tt] = attb[(size_t)l * 4096 + j];
  }
  const float* v1b = v1 + ((size_t)b * 4096 + (size_t)i * 64) * 256 + h * 32;
  for (int tt = tid; tt < 2048; tt += 256) {
    const int l = tt >> 5, d = tt & 31;
    s_v1[tt] = v1b[(size_t)l * 256 + d];
  }
  __syncthreads();

  const int j  = tid >> 2;
  const int d0 = (tid & 3) * 8;
  float acc[8];
  #pragma unroll
  for (int u = 0; u < 8; ++u) acc[u] = 0.f;

  const float* v2b =
      v2 + ((size_t)b * 4096 + j) * 256 + h * 32 + d0;  // + l*64*256
  for (int l = 0; l < 64; ++l) {
    const float* vp = v2b + (size_t)l * (64 * 256);
    if (l + 1 < 64) __builtin_prefetch(v2b + (size_t)(l + 1) * (64 * 256), 0, 0);
    const float a = s_att[l * 64 + j];
    const float4 x0 = *(const float4*)vp;
    const float4 x1 = *(const float4*)(vp + 4);
    const float* v1p = s_v1 + l * 32 + d0;
    acc[0] += a * v1p[0] * x0.x;
    acc[1] += a * v1p[1] * x0.y;
    acc[2] += a * v1p[2] * x0.z;
    acc[3] += a * v1p[3] * x0.w;
    acc[4] += a * v1p[4] * x1.x;
    acc[5] += a * v1p[5] * x1.y;
    acc[6] += a * v1p[6] * x1.z;
    acc[7] += a * v1p[7] * x1.w;
  }
  float* op = out_pre + ((size_t)b * 4096 + (size_t)i * 64 + j) * 256 + h * 32 + d0;
  #pragma unroll
  for (int u = 0; u < 8; ++u) op[u] = acc[u];
}

// ---------------------------------------------------------------------------
// K5: h_ln = LayerNorm(x_in + out_pre @ Wo) * g + b   (fused GEMM + LN)
// Block: 16 rows x 256 cols; LN per row with wave32 shuffle reduction.
// ---------------------------------------------------------------------------
__global__ void __launch_bounds__(256)
attn_out_ln_kernel(const float* __restrict__ out_pre,
                   const float* __restrict__ Wo, const float* __restrict__ x_in,
                   const float* __restrict__ g, const float* __restrict__ bta,
                   float* __restrict__ h_ln) {
  const int rowbase = blockIdx.x * 16;
  const int wid  = threadIdx.x >> 5;
  const int lane = threadIdx.x & 31;
  const int lrow = lane & 15;
  const int rh   = (lane >> 4) * 8;

  v8f acc[2] = {};
  gemm_panel<2>(out_pre, 256, rowbase, Wo, 256, wid * 32, 256, acc);

  __shared__ float s_h[16 * 256];
  #pragma unroll
  for (int t = 0; t < 2; ++t) {
    const int col = wid * 32 + t * 16 + lrow;
    #pragma unroll
    for (int r = 0; r < 8; ++r) {
      const int row = r + rh;
      s_h[row * 256 + col] =
          acc[t][r] + x_in[(size_t)(rowbase + row) * 256 + col];
    }
  }
  __syncthreads();

  #pragma unroll
  for (int rr = 0; rr < 2; ++rr) {
    const int row = wid * 2 + rr;
    float s = 0.f, sq = 0.f;
    #pragma unroll
    for (int u = 0; u < 8; ++u) {
      const float v = s_h[row * 256 + lane + u * 32];
      s += v;
      sq += v * v;
    }
    s  = wave_reduce_add(s);
    sq = wave_reduce_add(sq);
    const float mean = s * (1.0f / 256.0f);
    const float var  = sq * (1.0f / 256.0f) - mean * mean;
    const float rstd = rsqrtf(var + 1e-5f);
    #pragma unroll
    for (int u = 0; u < 8; ++u) {
      const int c = lane + u * 32;
      h_ln[(size_t)(rowbase + row) * 256 + c] =
          (s_h[row * 256 + c] - mean) * rstd * g[c] + bta[c];
    }
  }
}

// ---------------------------------------------------------------------------
// K6: out = LN( relu(h_ln@W1 + b1)@W2 + b2 + h_ln ) * g + b  (fully fused FFN)
// mid (16x512) kept in LDS with pitch 516 to avoid bank conflicts on the
// row-indexed A reads in step 2.
// ---------------------------------------------------------------------------
#define MID_PITCH 516
__global__ void __launch_bounds__(256)
ffn_kernel(const float* __restrict__ h_ln, const float* __restrict__ W1,
           const float* __restrict__ b1, const float* __restrict__ W2,
           const float* __restrict__ b2, const float* __restrict__ g,
           const float* __restrict__ bta, float* __restrict__ out) {
  const int rowbase = blockIdx.x * 16;
  const int wid   = threadIdx.x >> 5;
  const int lane  = threadIdx.x & 31;
  const int lrow  = lane & 15;
  const int khalf = lane >> 4;
  const int rh    = khalf * 8;

  __shared__ float s_mid[16 * MID_PITCH];  // relu(h_ln@W1 + b1), padded pitch
  __shared__ float s_y[16 * 256];          // final pre-LN rows

  // ---- step 1: mid = relu(h_ln @ W1 + b1), 4 tiles/wave over 512 cols ----
  v8f a1[4] = {};
  gemm_panel<4>(h_ln, 256, rowbase, W1, 512, wid * 64, 256, a1);
  #pragma unroll
  for (int t = 0; t < 4; ++t) {
    const int col = wid * 64 + t * 16 + lrow;
    const float bias = b1[col];
    #pragma unroll
    for (int r = 0; r < 8; ++r)
      s_mid[(r + rh) * MID_PITCH + col] = fmaxf(a1[t][r] + bias, 0.f);
  }
  __syncthreads();

  // ---- step 2: y = mid @ W2 + b2 + h_ln (A from LDS), 2 tiles/wave ----
  v8f a2[2] = {};
  {
    const float* arow = s_mid + lrow * MID_PITCH + 2 * khalf;
    for (int kb = 0; kb < 512; kb += 4) {
      v2f a = *(const v2f*)(arow + kb);  // ds_load_b64
      #pragma unroll
      for (int t = 0; t < 2; ++t) {
        const int col = wid * 32 + t * 16 + lrow;
        v2f bb;
        bb.x = W2[(size_t)(kb + 2 * khalf) * 256 + col];
        bb.y = W2[(size_t)(kb + 2 * khalf + 1) * 256 + col];
        a2[t] = wmma_f32x4(a, bb, a2[t]);
      }
    }
  }
  #pragma unroll
  for (int t = 0; t < 2; ++t) {
    const int col = wid * 32 + t * 16 + lrow;
    const float bias = b2[col];
    #pragma unroll
    for (int r = 0; r < 8; ++r) {
      const int row = r + rh;
      s_y[row * 256 + col] =
          a2[t][r] + bias + h_ln[(size_t)(rowbase + row) * 256 + col];
    }
  }
  __syncthreads();

  // ---- final LayerNorm ----
  #pragma unroll
  for (int rr = 0; rr < 2; ++rr) {
    const int row = wid * 2 + rr;
    float s = 0.f, sq = 0.f;
    #pragma unroll
    for (int u = 0; u < 8; ++u) {
      const float v = s_y[row * 256 + lane + u * 32];
      s += v;
      sq += v * v;
    }
    s  = wave_reduce_add(s);
    sq = wave_reduce_add(sq);
    const float mean = s * (1.0f / 256.0f);
    const float var  = sq * (1.0f / 256.0f) - mean * mean;
    const float rstd = rsqrtf(var + 1e-5f);
    #pragma unroll
    for (int u = 0; u < 8; ++u) {
      const int c = lane + u * 32;
      out[(size_t)(rowbase + row) * 256 + c] =
          (s_y[row * 256 + c] - mean) * rstd * g[c] + bta[c];
    }
  }
}

// ---------------------------------------------------------------------------
extern "C" void kernel_launch(void* const* d_in, const int* in_sizes, int n_in,
                              void* d_out, int out_size, void* d_ws,
                              size_t ws_size, hipStream_t stream) {
  const float* x_in   = (const float*)d_in[0];
  const float* Wq     = (const float*)d_in[1];
  const float* Wk     = (const float*)d_in[2];
  const float* Wv1    = (const float*)d_in[3];
  const float* Wv2    = (const float*)d_in[4];
  const float* Wo     = (const float*)d_in[5];
  const float* ln_a_g = (const float*)d_in[6];
  const float* ln_a_b = (const float*)d_in[7];
  const float* W1     = (const float*)d_in[8];
  const float* b1     = (const float*)d_in[9];
  const float* W2     = (const float*)d_in[10];
  const float* b2     = (const float*)d_in[11];
  const float* ln_o_g = (const float*)d_in[12];
  const float* ln_o_b = (const float*)d_in[13];

  // Workspace layout (floats). SZ = R*E = 8192*256 = 2,097,152 (8 MB).
  // q,k,v1,v2: 4*SZ ; att: 2*SZ (B*H*N^3 = 4,194,304) ; out_pre: SZ ; h_ln: SZ
  // Total 8*SZ floats = 64 MB.
  const size_t SZ = 2097152;
  float* ws      = (float*)d_ws;
  float* q       = ws;
  float* k       = ws + SZ;
  float* v1      = ws + 2 * SZ;
  float* v2      = ws + 3 * SZ;
  float* att     = ws + 4 * SZ;  // occupies [4*SZ, 6*SZ)
  float* out_pre = ws + 6 * SZ;
  float* h_ln    = ws + 7 * SZ;

  proj_kernel<<<dim3(512, 4), 256, 0, stream>>>(x_in, Wq, Wk, Wv1, Wv2, q, k,
                                                v1, v2);
  scores_kernel<<<1024, 256, 0, stream>>>(q, k, att);
  softmax_l_kernel<<<256, 256, 0, stream>>>(att);
  edge_av_kernel<<<1024, 256, 0, stream>>>(att, v1, v2, out_pre);
  attn_out_ln_kernel<<<512, 256, 0, stream>>>(out_pre, Wo, x_in, ln_a_g,
                                              ln_a_b, h_ln);
  ffn_kernel<<<512, 256, 0, stream>>>(h_ln, W1, b1, W2, b2, ln_o_g, ln_o_b,
                                      (float*)d_out);
}